// TMQI_74612171866769
// MI455X (gfx1250) — compile-verified
//
#include <hip/hip_runtime.h>
#include <cmath>

typedef __attribute__((ext_vector_type(2))) float v2f;
typedef __attribute__((ext_vector_type(8))) float v8f;

struct GW { float w[11]; };

__device__ __forceinline__ int imin_i(int a, int b) { return a < b ? a : b; }

// Accumulator layout (floats): [0..39] slsum[level][b], [40..47] hdrmin,
// [48..55] hdrmax, [56..63] ldrsum, [64..71] sigsum
__global__ void tmqi_init(float* acc) {
    int t = blockIdx.x * blockDim.x + threadIdx.x;
    if (t < 72) acc[t] = 0.0f;
}

__global__ void tmqi_rgb2y(const float* __restrict__ hdr, const float* __restrict__ ldr,
                           float* __restrict__ lh, float* __restrict__ ll,
                           int HW, int total) {
    int i = blockIdx.x * blockDim.x + threadIdx.x;
    if (i >= total) return;
    int b = i / HW;
    int p = i - b * HW;
    const float* h = hdr + (size_t)b * 3 * HW;
    const float* l = ldr + (size_t)b * 3 * HW;
    lh[i] = 0.2126f * h[p] + 0.7152f * h[p + HW] + 0.0722f * h[p + 2 * HW];
    ll[i] = 0.2126f * l[p] + 0.7152f * l[p + HW] + 0.0722f * l[p + 2 * HW];
}

// One block per image: hdr min/max + ldr sum.
__global__ __launch_bounds__(256)
void tmqi_reduce(const float* __restrict__ lh, const float* __restrict__ ll,
                 float* __restrict__ acc, int HW) {
    int b = blockIdx.x;
    int t = threadIdx.x;
    const float* ph = lh + (size_t)b * HW;
    const float* pl = ll + (size_t)b * HW;
    float mn = 3.4e38f, mx = -3.4e38f, sm = 0.0f;
    for (int i = t; i < HW; i += 256) {
        float v = ph[i];
        mn = fminf(mn, v);
        mx = fmaxf(mx, v);
        sm += pl[i];
    }
    __shared__ float smn[256], smx[256], ssm[256];
    smn[t] = mn; smx[t] = mx; ssm[t] = sm;
    __syncthreads();
    for (int o = 128; o > 0; o >>= 1) {
        if (t < o) {
            smn[t] = fminf(smn[t], smn[t + o]);
            smx[t] = fmaxf(smx[t], smx[t + o]);
            ssm[t] += ssm[t + o];
        }
        __syncthreads();
    }
    if (t == 0) {
        acc[40 + b] = smn[0];
        acc[48 + b] = smx[0];
        acc[56 + b] = ssm[0];
    }
}

__global__ void tmqi_normalize(float* __restrict__ lh, const float* __restrict__ acc,
                               int HW, int total) {
    int i = blockIdx.x * blockDim.x + threadIdx.x;
    if (i >= total) return;
    int b = i / HW;
    float mn = acc[40 + b], mx = acc[48 + b];
    lh[i] = 4294967295.0f * (lh[i] - mn) / (mx - mn);
}

// One thread per 11x11 window: sigma of window, summed per image.
__global__ void tmqi_sigwin(const float* __restrict__ ll, float* __restrict__ acc, int H) {
    int NW = H / 11;
    int i = blockIdx.x * blockDim.x + threadIdx.x;
    if (i >= 8 * NW * NW) return;
    int b = i / (NW * NW);
    int r = i - b * NW * NW;
    int wy = r / NW, wx = r - wy * NW;
    const float* p = ll + (size_t)b * H * H + (size_t)(wy * 11) * H + wx * 11;
    float s = 0.0f, s2 = 0.0f;
    for (int y = 0; y < 11; y++)
        for (int x = 0; x < 11; x++) {
            float v = p[(size_t)y * H + x];
            s += v; s2 += v * v;
        }
    float m = s * (1.0f / 121.0f);
    float var = s2 * (1.0f / 121.0f) - m * m;
    atomicAdd(&acc[64 + b], sqrtf(var + 1e-5f));
}

// 2x2 average-pool both pyramids.
__global__ void tmqi_down2(const float* __restrict__ ih, float* __restrict__ oh,
                           const float* __restrict__ il, float* __restrict__ ol,
                           int Hin, int total) {
    int Ho = Hin >> 1;
    int i = blockIdx.x * blockDim.x + threadIdx.x;
    if (i >= total) return;
    int b = i / (Ho * Ho);
    int r = i - b * Ho * Ho;
    int y = r / Ho, x = r - y * Ho;
    size_t base = (size_t)b * Hin * Hin + (size_t)(2 * y) * Hin + 2 * x;
    oh[i] = 0.25f * (ih[base] + ih[base + 1] + ih[base + Hin] + ih[base + Hin + 1]);
    ol[i] = 0.25f * (il[base] + il[base + 1] + il[base + Hin] + il[base + Hin + 1]);
}

// Fused _slocal for one pyramid level. One wave (32 lanes) per 16x16 output tile.
// Horizontal 11-tap pass (5 channels) in VALU -> LDS; vertical 11-tap pass as a
// banded GEMM: Out[16x16] = Band[16x28] x H[28x16] via 7 chained
// V_WMMA_F32_16X16X4_F32 per channel (f32-exact, matches reference math).
__global__ __launch_bounds__(32)
void tmqi_slocal(const float* __restrict__ hdr, const float* __restrict__ ldr,
                 int H, int outH, float* __restrict__ slsum, int level,
                 GW gw, float uu, float inv) {
    const int tx = blockIdx.x, ty = blockIdx.y, b = blockIdx.z;
    const int lane = threadIdx.x;
    const float* i1 = hdr + (size_t)b * H * H;
    const float* i2 = ldr + (size_t)b * H * H;
    const int x0 = tx * 16, y0 = ty * 16;

    __shared__ float s1[26][28];
    __shared__ float s2[26][28];
    __shared__ float Hc[5][28][16];
    // Zero-padded vertical weight table: wext[15+d] = w[d] for d in [0,11),
    // zeros elsewhere; valid for indices (j - m) in [-15, 27].
    __shared__ float wext[44];

    // Fill padded weight table (static indices -> scalar kernarg loads only).
    for (int d = lane; d < 44; d += 32) wext[d] = 0.0f;
    __syncthreads();
    if (lane == 0) {
        wext[15] = gw.w[0];  wext[16] = gw.w[1];  wext[17] = gw.w[2];
        wext[18] = gw.w[3];  wext[19] = gw.w[4];  wext[20] = gw.w[5];
        wext[21] = gw.w[6];  wext[22] = gw.w[7];  wext[23] = gw.w[8];
        wext[24] = gw.w[9];  wext[25] = gw.w[10];
    }

    // Load 26x26 halo tile (clamped; clamps only feed masked-out outputs).
    for (int idx = lane; idx < 26 * 26; idx += 32) {
        int r = idx / 26, c = idx - r * 26;
        int yy = imin_i(y0 + r, H - 1);
        int xx = imin_i(x0 + c, H - 1);
        s1[r][c] = i1[(size_t)yy * H + xx];
        s2[r][c] = i2[(size_t)yy * H + xx];
    }
    __syncthreads();

    // Horizontal pass: 26 rows x 16 cols x 5 channels.
    for (int idx = lane; idx < 26 * 16; idx += 32) {
        int r = idx >> 4, c = idx & 15;
        float a0 = 0.f, a1 = 0.f, a2 = 0.f, a3 = 0.f, a4 = 0.f;
#pragma unroll
        for (int k = 0; k < 11; k++) {
            float wv = gw.w[k];
            float v1 = s1[r][c + k];
            float v2 = s2[r][c + k];
            a0 += wv * v1;
            a1 += wv * v2;
            a2 += wv * v1 * v1;
            a3 += wv * v2 * v2;
            a4 += wv * v1 * v2;
        }
        Hc[0][r][c] = a0; Hc[1][r][c] = a1; Hc[2][r][c] = a2;
        Hc[3][r][c] = a3; Hc[4][r][c] = a4;
    }
    // Zero K-pad rows 26,27.
    {
        int r = 26 + (lane >> 4), c = lane & 15;
#pragma unroll
        for (int ch = 0; ch < 5; ch++) Hc[ch][r][c] = 0.0f;
    }
    __syncthreads();

    // Fragment indices (ISA 7.12.2 layouts, wave32).
    const int m = lane & 15;        // A: output row; B/D: column N
    const int half = lane >> 4;
    const int kb = half * 2;        // K sub-index within a 4-wide chunk

    // Precompute all 7 band-matrix A fragments once (channel-invariant):
    // A[m][j] = w[j - m] for j - m in [0, 11), via the padded LDS table.
    v2f afrag[7];
#pragma unroll
    for (int q = 0; q < 7; q++) {
        int j0 = 4 * q + kb;
        afrag[q].x = wext[15 + j0 - m];
        afrag[q].y = wext[16 + j0 - m];
    }

    // Vertical pass via WMMA f32 16x16x4, K = 28 in 7 chunks per channel.
    v8f accv[5];
#pragma unroll
    for (int ch = 0; ch < 5; ch++) {
        const float* hrow = &Hc[ch][0][0];
        v8f c = {};
#pragma unroll
        for (int q = 0; q < 7; q++) {
            int j0 = 4 * q + kb;
            v2f bb;
            bb.x = hrow[j0 * 16 + m];
            bb.y = hrow[(j0 + 1) * 16 + m];
            c = __builtin_amdgcn_wmma_f32_16x16x4_f32(
                    false, afrag[q], false, bb, (short)0, c, false, false);
        }
        accv[ch] = c;
    }

    // Epilogue: s_map per element, masked partial sum.
    float partial = 0.0f;
#pragma unroll
    for (int v = 0; v < 8; v++) {
        int oy = y0 + v + 8 * half;   // D layout: VGPR v holds rows v / v+8
        int ox = x0 + m;
        float mu1 = accv[0][v], mu2 = accv[1][v];
        float s1sq = accv[2][v] - mu1 * mu1;
        float s2sq = accv[3][v] - mu2 * mu2;
        float s12  = accv[4][v] - mu1 * mu2;
        float sg1 = s1sq > 0.0f ? sqrtf(s1sq) : 0.0f;
        float sg2 = s2sq > 0.0f ? sqrtf(s2sq) : 0.0f;
        float p1 = 0.5f * (1.0f + erff((sg1 - uu) * inv));
        float p2 = 0.5f * (1.0f + erff((sg2 - uu) * inv));
        float smap = (2.0f * p1 * p2 + 0.01f) / (p1 * p1 + p2 * p2 + 0.01f)
                   * (s12 + 10.0f) / (sg1 * sg2 + 10.0f);
        if (oy < outH && ox < outH) partial += smap;
    }

    __shared__ float red[32];
    red[lane] = partial;
    __syncthreads();
    if (lane == 0) {
        float t = 0.0f;
        for (int i = 0; i < 32; i++) t += red[i];
        atomicAdd(&slsum[level * 8 + b], t);
    }
}

__global__ void tmqi_final(const float* __restrict__ acc, float* __restrict__ out, int H) {
    int b = threadIdx.x;
    __shared__ float vals[8];
    if (b < 8) {
        float HWf = (float)H * (float)H;
        int nw = H / 11;
        float u = acc[56 + b] / HWf;
        float sig = acc[64 + b] / (float)(nw * nw);
        float x = sig / 64.29f;
        float cc = 5009.72154f * powf(x, 3.4f) * powf(1.0f - x, 9.1f);
        float pc = cc / 3.3323f;
        float z = (u - 115.94f) / 27.99f;
        float bb = (0.39894f / 27.99f) * expf(-0.5f * z * z);
        float pb = bb / 0.014253f;
        float n = pb * pc;
        const float wgt[5] = {0.0448f, 0.2856f, 0.3001f, 0.2363f, 0.1333f};
        float s = 1.0f;
        for (int l = 0; l < 5; l++) {
            int hl = H >> l;
            float ol = (float)(hl - 10);
            float sl = acc[l * 8 + b] / (ol * ol);
            s *= powf(sl, wgt[l]);
        }
        float q = 0.8012f * powf(s, 0.3046f) + (1.0f - 0.8012f) * powf(n, 0.7088f);
        vals[b] = 1.0f - q;
    }
    __syncthreads();
    if (b == 0) {
        float t = 0.0f;
        for (int i = 0; i < 8; i++) t += vals[i];
        out[0] = t / 8.0f;
    }
}

extern "C" void kernel_launch(void* const* d_in, const int* in_sizes, int n_in,
                              void* d_out, int out_size, void* d_ws, size_t ws_size,
                              hipStream_t stream) {
    (void)in_sizes; (void)n_in; (void)out_size; (void)ws_size;
    const int B = 8, H = 704, HW = H * H;
    const float* hdr = (const float*)d_in[0];
    const float* ldr = (const float*)d_in[1];
    float* outp = (float*)d_out;
    float* ws = (float*)d_ws;

    // Pyramid offsets (floats).
    size_t off[5];
    size_t P = 0;
    for (int l = 0; l < 5; l++) {
        off[l] = P;
        size_t hl = (size_t)(H >> l);
        P += (size_t)B * hl * hl;
    }
    float* lh = ws;          // hdr pyramid
    float* ll = ws + P;      // ldr pyramid
    float* acc = ws + 2 * P; // 72 accumulator floats

    // Host-side constants: normalized 1-D Gaussian, per-level CSF erf params.
    GW gw;
    {
        double g[11], s = 0.0;
        for (int i = 0; i < 11; i++) {
            double d = (i - 5.0) / 1.5;
            g[i] = std::exp(-0.5 * d * d);
            s += g[i];
        }
        for (int i = 0; i < 11; i++) gw.w[i] = (float)(g[i] / s);
    }
    float uu[5], inv[5];
    {
        double sf = 32.0;
        for (int l = 0; l < 5; l++) {
            sf *= 0.5; // 16, 8, 4, 2, 1
            double csf = 100.0 * 2.6 * (0.0192 + 0.114 * sf)
                       * std::exp(-std::pow(0.114 * sf, 1.1));
            double u = 128.0 / (1.4 * csf);
            double sg = u / 3.0;
            uu[l]  = (float)u;
            inv[l] = (float)(1.0 / (sg * std::sqrt(2.0)));
        }
    }

    tmqi_init<<<1, 128, 0, stream>>>(acc);

    int total0 = B * HW;
    tmqi_rgb2y<<<(total0 + 255) / 256, 256, 0, stream>>>(hdr, ldr, lh, ll, HW, total0);
    tmqi_reduce<<<B, 256, 0, stream>>>(lh, ll, acc, HW);
    tmqi_normalize<<<(total0 + 255) / 256, 256, 0, stream>>>(lh, acc, HW, total0);

    int NW = H / 11;
    int nsig = B * NW * NW;
    tmqi_sigwin<<<(nsig + 255) / 256, 256, 0, stream>>>(ll, acc, H);

    for (int l = 0; l < 5; l++) {
        int Hl = H >> l;
        int outl = Hl - 10;
        int tiles = (outl + 15) / 16;
        dim3 grid(tiles, tiles, B);
        tmqi_slocal<<<grid, 32, 0, stream>>>(lh + off[l], ll + off[l],
                                             Hl, outl, acc, l, gw, uu[l], inv[l]);
        if (l < 4) {
            int Ho = Hl >> 1;
            int tot = B * Ho * Ho;
            tmqi_down2<<<(tot + 255) / 256, 256, 0, stream>>>(
                lh + off[l], lh + off[l + 1], ll + off[l], ll + off[l + 1], Hl, tot);
        }
    }

    tmqi_final<<<1, 32, 0, stream>>>(acc, outp, H);
}